// DeepND_ST_29033978921059
// MI455X (gfx1250) — compile-verified
//
#include <hip/hip_runtime.h>
#include <math.h>

#define NN 100000
#define KK 4
#define EE 1600000
#define EPSV 1e-05f

typedef float v2f __attribute__((ext_vector_type(2)));
typedef float v8f __attribute__((ext_vector_type(8)));

// LDS row stride for the A tile: 260 ≡ 4 (mod 64) makes the per-lane dword
// addresses m*260 + kk + 2*half cover 64 distinct banks -> conflict-free
// ds_load_b64 for the WMMA A fragment.
#define A_STRIDE 260

// ---------------------------------------------------------------------------
// GEMM1: H[N x 64] = X[N x 256] @ W[256 x 64], fp32 via V_WMMA_F32_16X16X4_F32
// Block = 128 threads (4 waves). blockIdx.x = 16-row tile (N = 6250*16 exact).
// The 16x256 A tile is staged once per block in LDS (shared by all 4 waves,
// removing the 4x redundant L2 reads of `flatten`); wave w computes the
// 16x16 output tile at columns [16w, 16w+16).
// ---------------------------------------------------------------------------
__global__ void __launch_bounds__(128)
gemm1_wmma_f32(const float* __restrict__ X, const float* __restrict__ W,
               float* __restrict__ H) {
  __shared__ float As[16 * A_STRIDE];

  const int lane = threadIdx.x & 31;
  const int wave = threadIdx.x >> 5;          // column tile 0..3
  const int m    = lane & 15;                 // row (A/D) or col (B/D) index
  const int half = lane >> 4;                 // 0: K={0,1}, 1: K={2,3}
  const int row0 = blockIdx.x * 16;
  const int col  = wave * 16 + m;             // B/D column for this lane

  // Cooperative stage of A tile: 16 rows x 256 cols = 1024 float4 chunks.
  for (int i = threadIdx.x; i < 16 * 64; i += 128) {
    const int r  = i >> 6;                    // 64 float4s per row
    const int c4 = i & 63;
    float4 v = *(const float4*)(X + (size_t)(row0 + r) * 256 + c4 * 4);
    *(float4*)(&As[r * A_STRIDE + c4 * 4]) = v;
  }
  __syncthreads();

  const float* __restrict__ arow = &As[m * A_STRIDE];
  v8f acc = {};
#pragma unroll 4
  for (int kk = 0; kk < 256; kk += 4) {
    const int k0 = kk + half * 2;
    v2f a = *(const v2f*)(arow + k0);                       // A[m][k0..k0+1] (LDS)
    v2f b;
    b.x = W[(size_t)k0 * 64 + col];                         // B[k0][col]
    b.y = W[(size_t)(k0 + 1) * 64 + col];                   // B[k0+1][col]
    acc = __builtin_amdgcn_wmma_f32_16x16x4_f32(
        /*neg_a=*/false, a, /*neg_b=*/false, b,
        /*c_mod=*/(short)0, acc, /*reuse_a=*/false, /*reuse_b=*/false);
  }
#pragma unroll
  for (int r = 0; r < 8; ++r) {
    const int mrow = row0 + r + 8 * half;                   // D row for VGPR r
    H[(size_t)mrow * 64 + col] = acc[r];
  }
}

// ---------------------------------------------------------------------------
// Degree: deg[dst[e]] += 1  (float atomics, deg pre-zeroed)
// ---------------------------------------------------------------------------
__global__ void edge_deg(const int* __restrict__ dst, float* __restrict__ deg,
                         int E) {
  int i = blockIdx.x * blockDim.x + threadIdx.x;
  if (i < E) atomicAdd(&deg[dst[i]], 1.0f);
}

// dis = rsqrt(deg + 2), in place
__global__ void dis_finish(float* __restrict__ d, int n) {
  int i = blockIdx.x * blockDim.x + threadIdx.x;
  if (i < n) d[i] = rsqrtf(d[i] + 2.0f);
}

// ---------------------------------------------------------------------------
// Conv1 edge aggregation: 16 lanes per edge, 4 channels each (float4 gather,
// 4 x global_atomic_add_f32 scatter). h / agg are L2-resident (25.6 MB each).
// ---------------------------------------------------------------------------
__global__ void edge_agg64(const int* __restrict__ src,
                           const int* __restrict__ dst,
                           const float* __restrict__ dis,
                           const float* __restrict__ h,
                           float* __restrict__ agg, int E) {
  int t = blockIdx.x * blockDim.x + threadIdx.x;
  int e = t >> 4, g = t & 15;
  if (e >= E) return;
  int s = src[e], d = dst[e];
  float coef = dis[s] * dis[d];
  float4 hv = *(const float4*)(h + (size_t)s * 64 + g * 4);
  float* out = agg + (size_t)d * 64 + g * 4;
  atomicAdd(out + 0, hv.x * coef);
  atomicAdd(out + 1, hv.y * coef);
  atomicAdd(out + 2, hv.z * coef);
  atomicAdd(out + 3, hv.w * coef);
}

// x = relu(agg + (2*dis^2)*h + b1), written in place into agg
__global__ void post_conv1(const float* __restrict__ h,
                           float* __restrict__ x,
                           const float* __restrict__ dis,
                           const float* __restrict__ b1, int n) {
  int t = blockIdx.x * blockDim.x + threadIdx.x;
  int node = t >> 4, g = t & 15;
  if (node >= n) return;
  float dd = dis[node];
  float c = 2.0f * dd * dd;
  size_t base = (size_t)node * 64 + g * 4;
  float4 hv = *(const float4*)(h + base);
  float4 av = *(const float4*)(x + base);
  float4 bv = *(const float4*)(b1 + g * 4);
  float4 r;
  r.x = fmaxf(av.x + c * hv.x + bv.x, 0.0f);
  r.y = fmaxf(av.y + c * hv.y + bv.y, 0.0f);
  r.z = fmaxf(av.z + c * hv.z + bv.z, 0.0f);
  r.w = fmaxf(av.w + c * hv.w + bv.w, 0.0f);
  *(float4*)(x + base) = r;
}

// ---------------------------------------------------------------------------
// Per-channel sum / sum-of-squares over all N rows (stats[0..63]=sum,
// stats[64..127]=sumsq, pre-zeroed). Block = 256 = 4 row-subsets x 64 channels.
// ---------------------------------------------------------------------------
__global__ void __launch_bounds__(256)
stats_reduce(const float* __restrict__ x, float* __restrict__ stats, int n) {
  __shared__ float ssum[64];
  __shared__ float ssq[64];
  int c = threadIdx.x & 63;
  int sub = threadIdx.x >> 6;       // 0..3
  int base = blockIdx.x * 256;      // 256 rows per block
  float s = 0.0f, q = 0.0f;
  for (int r = base + sub; r < base + 256 && r < n; r += 4) {
    float v = x[(size_t)r * 64 + c];
    s += v;
    q += v * v;
  }
  if (threadIdx.x < 64) { ssum[threadIdx.x] = 0.0f; ssq[threadIdx.x] = 0.0f; }
  __syncthreads();
  atomicAdd(&ssum[c], s);
  atomicAdd(&ssq[c], q);
  __syncthreads();
  if (threadIdx.x < 64) {
    atomicAdd(&stats[c], ssum[c]);
    atomicAdd(&stats[64 + c], ssq[c]);
  }
}

// stats -> mu (0..63), rstd (64..127)
__global__ void stats_final(float* __restrict__ stats, int n) {
  int c = threadIdx.x;
  if (c >= 64) return;
  float mu = stats[c] / (float)n;
  float var = stats[64 + c] / (float)n - mu * mu;
  stats[c] = mu;
  stats[64 + c] = rsqrtf(var + EPSV);
}

// ---------------------------------------------------------------------------
// Normalize + GEMM2 (64 -> 2), one thread per node.
// h2[n][j] = sum_c ((x[n][c]-mu)*rstd*gamma+beta) * W2[c][j]
// ---------------------------------------------------------------------------
__global__ void norm_gemm2(const float* __restrict__ x,
                           const float* __restrict__ stats,
                           const float* __restrict__ gmm,
                           const float* __restrict__ bta,
                           const float* __restrict__ W2,
                           float* __restrict__ h2, int n) {
  int node = blockIdx.x * blockDim.x + threadIdx.x;
  if (node >= n) return;
  const float* xr = x + (size_t)node * 64;
  float o0 = 0.0f, o1 = 0.0f;
#pragma unroll 8
  for (int c = 0; c < 64; ++c) {
    float xn = (xr[c] - stats[c]) * stats[64 + c] * gmm[c] + bta[c];
    o0 += xn * W2[c * 2 + 0];
    o1 += xn * W2[c * 2 + 1];
  }
  h2[(size_t)node * 2 + 0] = o0;
  h2[(size_t)node * 2 + 1] = o1;
}

// Conv2 edge aggregation: one thread per edge (2 channels)
__global__ void edge_agg2(const int* __restrict__ src,
                          const int* __restrict__ dst,
                          const float* __restrict__ dis,
                          const float* __restrict__ h2,
                          float* __restrict__ agg2, int E) {
  int e = blockIdx.x * blockDim.x + threadIdx.x;
  if (e >= E) return;
  int s = src[e], d = dst[e];
  float coef = dis[s] * dis[d];
  float2 hv = *(const float2*)(h2 + (size_t)s * 2);
  atomicAdd(agg2 + (size_t)d * 2 + 0, hv.x * coef);
  atomicAdd(agg2 + (size_t)d * 2 + 1, hv.y * coef);
}

// logits[k][n][j] = agg2 + (2*dis^2)*h2 + b2
__global__ void logits_finish(const float* __restrict__ agg2,
                              const float* __restrict__ dis,
                              const float* __restrict__ h2,
                              const float* __restrict__ b2,
                              float* __restrict__ logits, int n) {
  int node = blockIdx.x * blockDim.x + threadIdx.x;
  if (node >= n) return;
  float dd = dis[node];
  float c = 2.0f * dd * dd;
  size_t b = (size_t)node * 2;
  logits[b + 0] = agg2[b + 0] + c * h2[b + 0] + b2[0];
  logits[b + 1] = agg2[b + 1] + c * h2[b + 1] + b2[1];
}

// ---------------------------------------------------------------------------
// Combine: gate softmax(moe @ Wg + bg), weighted log-softmax / softmax of the
// 2-class expert logits. out[0:2N) = results_concat, out[2N:4N) = probs.
// ---------------------------------------------------------------------------
__global__ void combine(const float* __restrict__ moe,
                        const float* __restrict__ Wg,
                        const float* __restrict__ bg,
                        const float* __restrict__ logits,
                        float* __restrict__ out, int n) {
  int node = blockIdx.x * blockDim.x + threadIdx.x;
  if (node >= n) return;
  float acc[KK] = {bg[0], bg[1], bg[2], bg[3]};
  const float* mr = moe + (size_t)node * 256;
#pragma unroll 4
  for (int c = 0; c < 256; c += 4) {
    float4 mv = *(const float4*)(mr + c);
    const float* wr = Wg + (size_t)c * KK;
#pragma unroll
    for (int j = 0; j < KK; ++j)
      acc[j] += mv.x * wr[j] + mv.y * wr[KK + j] + mv.z * wr[2 * KK + j] +
                mv.w * wr[3 * KK + j];
  }
  float gm = fmaxf(fmaxf(acc[0], acc[1]), fmaxf(acc[2], acc[3]));
  float gs = 0.0f;
#pragma unroll
  for (int j = 0; j < KK; ++j) { acc[j] = expf(acc[j] - gm); gs += acc[j]; }
  float inv = 1.0f / gs;

  float lp0 = 0.0f, lp1 = 0.0f, p0 = 0.0f, p1 = 0.0f;
#pragma unroll
  for (int k = 0; k < KK; ++k) {
    float w = acc[k] * inv;
    float l0 = logits[((size_t)k * n + node) * 2 + 0];
    float l1 = logits[((size_t)k * n + node) * 2 + 1];
    float m = fmaxf(l0, l1);
    float e0 = expf(l0 - m), e1 = expf(l1 - m);
    float lse = m + logf(e0 + e1);
    float a0 = l0 - lse, a1 = l1 - lse;
    lp0 += w * a0;  lp1 += w * a1;
    p0  += w * expf(a0);  p1 += w * expf(a1);
  }
  out[(size_t)node * 2 + 0] = lp0;
  out[(size_t)node * 2 + 1] = lp1;
  out[(size_t)2 * n + (size_t)node * 2 + 0] = p0;
  out[(size_t)2 * n + (size_t)node * 2 + 1] = p1;
}

// ---------------------------------------------------------------------------
extern "C" void kernel_launch(void* const* d_in, const int* in_sizes, int n_in,
                              void* d_out, int out_size, void* d_ws,
                              size_t ws_size, hipStream_t stream) {
  const float* flatten = (const float*)d_in[0];   // N x 256
  const float* moe     = (const float*)d_in[1];   // N x 256
  const int*   eidx    = (const int*)d_in[2];     // K x 2 x E
  const float* W1      = (const float*)d_in[3];   // K x 256 x 64
  const float* b1      = (const float*)d_in[4];   // K x 64
  const float* gmm     = (const float*)d_in[5];   // K x 64
  const float* bta     = (const float*)d_in[6];   // K x 64
  const float* W2      = (const float*)d_in[7];   // K x 64 x 2
  const float* b2      = (const float*)d_in[8];   // K x 2
  const float* Wg      = (const float*)d_in[9];   // 256 x 4
  const float* bg      = (const float*)d_in[10];  // 4
  float* out = (float*)d_out;

  // Workspace layout (floats); total ~14.1M floats = 56.4 MB
  float* ws     = (float*)d_ws;
  float* h      = ws;                         // N*64
  float* x      = h  + (size_t)NN * 64;       // N*64 (agg, then post-relu conv1 x)
  float* dis    = x  + (size_t)NN * 64;       // N (deg, then rsqrt)
  float* stats  = dis + NN;                   // 128
  float* h2     = stats + 128;                // N*2
  float* agg2   = h2 + (size_t)NN * 2;        // N*2
  float* logits = agg2 + (size_t)NN * 2;      // K*N*2

  const int TB = 256;
  const int gN   = (NN + TB - 1) / TB;            // per-node grids
  const int gN16 = (NN * 16 + TB - 1) / TB;       // node x 16 grids
  const int gE   = (EE + TB - 1) / TB;            // per-edge grids
  const int gE16 = (EE / 16);                     // edge x 16 grid: E*16/256

  for (int k = 0; k < KK; ++k) {
    const int* srcK = eidx + (size_t)k * 2 * EE;
    const int* dstK = srcK + EE;

    // GEMM1 via fp32 WMMA (A tile staged in LDS)
    gemm1_wmma_f32<<<NN / 16, 128, 0, stream>>>(flatten, W1 + (size_t)k * 256 * 64, h);

    // degree -> dis (shared by both convs of this expert)
    hipMemsetAsync(dis, 0, sizeof(float) * NN, stream);
    edge_deg<<<gE, TB, 0, stream>>>(dstK, dis, EE);
    dis_finish<<<gN, TB, 0, stream>>>(dis, NN);

    // conv1 aggregation
    hipMemsetAsync(x, 0, sizeof(float) * (size_t)NN * 64, stream);
    edge_agg64<<<gE16, TB, 0, stream>>>(srcK, dstK, dis, h, x, EE);
    post_conv1<<<gN16, TB, 0, stream>>>(h, x, dis, b1 + k * 64, NN);

    // per-channel mean/var -> mu, rstd
    hipMemsetAsync(stats, 0, sizeof(float) * 128, stream);
    stats_reduce<<<gN, TB, 0, stream>>>(x, stats, NN);
    stats_final<<<1, 64, 0, stream>>>(stats, NN);

    // normalize + GEMM2 (64 -> 2)
    norm_gemm2<<<gN, TB, 0, stream>>>(x, stats, gmm + k * 64, bta + k * 64,
                                      W2 + (size_t)k * 128, h2, NN);

    // conv2 aggregation + logits
    hipMemsetAsync(agg2, 0, sizeof(float) * (size_t)NN * 2, stream);
    edge_agg2<<<gE, TB, 0, stream>>>(srcK, dstK, dis, h2, agg2, EE);
    logits_finish<<<gN, TB, 0, stream>>>(agg2, dis, h2, b2 + k * 2,
                                         logits + (size_t)k * NN * 2, NN);
  }

  combine<<<gN, TB, 0, stream>>>(moe, Wg, bg, logits, out, NN);
}